// GCN_40450001993771
// MI455X (gfx1250) — compile-verified
//
#include <hip/hip_runtime.h>
#include <hip/hip_bf16.h>

// ---------------- problem constants (from reference) ----------------
#define NN   50000
#define EE   800000
#define GG   64
#define DIN  96
#define DH   96
#define DOUT 10
#define MD   30

typedef __attribute__((ext_vector_type(2))) float v2f;
typedef __attribute__((ext_vector_type(8))) float v8f;

// ---------------- small utility kernels ----------------
__global__ void zero_f32(float* __restrict__ p, int n) {
    int i = blockIdx.x * blockDim.x + threadIdx.x;
    if (i < n) p[i] = 0.0f;
}

__global__ void pool_init(float* __restrict__ pooled, float* __restrict__ cnts,
                          int* __restrict__ first) {
    int i = blockIdx.x * blockDim.x + threadIdx.x;
    if (i < GG * DH) pooled[i] = 0.0f;
    if (i < GG) { cnts[i] = 0.0f; first[i] = 0x7fffffff; }
}

// deg[dst] += w  (self-loop +1 added in rsqrt kernel)
__global__ void deg_kernel(const int* __restrict__ dst, const float* __restrict__ w,
                           float* __restrict__ deg, int e) {
    int i = blockIdx.x * blockDim.x + threadIdx.x;
    if (i < e) atomicAdd(&deg[dst[i]], w[i]);
}

__global__ void rsqrt_kernel(float* __restrict__ deg_to_dinv, int n) {
    int i = blockIdx.x * blockDim.x + threadIdx.x;
    if (i < n) deg_to_dinv[i] = rsqrtf(deg_to_dinv[i] + 1.0f);
}

// ---------------- WMMA GEMM: H[nrows,96] = X[nrows,96] @ W[96,96] ----------------
// One wave computes a 16-row tile x all 96 cols. f32 WMMA 16x16x4.
// A 16x4 layout: lane&15 = M, (lane>>4)*2 + vgpr = K.
// B 4x16 layout: lane&15 = N, (lane>>4)*2 + vgpr = K.
// C/D 16x16:     vgpr r -> M = r + (lane>>4)*8, N = lane&15.
__global__ void gemm96_wmma(const float* __restrict__ X, const float* __restrict__ Wmat,
                            float* __restrict__ H, int nrows) {
    __shared__ float sW[96 * 96];
    for (int i = threadIdx.x; i < 96 * 96; i += blockDim.x) sW[i] = Wmat[i];
    __syncthreads();

    const int wave = threadIdx.x >> 5;
    const int lane = threadIdx.x & 31;
    const int tile = blockIdx.x * (blockDim.x >> 5) + wave;
    const int row0 = tile * 16;
    if (row0 >= nrows) return;   // wave-uniform: EXEC stays all-ones for WMMA

    const int m    = lane & 15;
    const int half = lane >> 4;

    v8f acc[6] = {};
    const float* xrow = X + (size_t)(row0 + m) * 96;

    for (int k0 = 0; k0 < 96; k0 += 4) {
        const int kk = k0 + half * 2;
        v2f a = *(const v2f*)(xrow + kk);          // contiguous pair, 8B aligned
#pragma unroll
        for (int nt = 0; nt < 6; ++nt) {
            v2f b;
            b.x = sW[(kk + 0) * 96 + nt * 16 + m];
            b.y = sW[(kk + 1) * 96 + nt * 16 + m];
            acc[nt] = __builtin_amdgcn_wmma_f32_16x16x4_f32(
                false, a, false, b, (short)0, acc[nt], false, false);
        }
    }

#pragma unroll
    for (int nt = 0; nt < 6; ++nt) {
#pragma unroll
        for (int r = 0; r < 8; ++r) {
            const int row = row0 + r + half * 8;
            H[(size_t)row * 96 + nt * 16 + m] = acc[nt][r];
        }
    }
}

// ---------------- edge scatter: agg[dst] += h[src] * norm ----------------
// thread = (edge, 8-feature chunk); 12 chunks of 8 cover DH=96.
__global__ void edge_agg(const int* __restrict__ src, const int* __restrict__ dst,
                         const float* __restrict__ w, const float* __restrict__ dinv,
                         const float* __restrict__ h, float* __restrict__ agg) {
    int tid = blockIdx.x * blockDim.x + threadIdx.x;
    if (tid >= EE * 12) return;
    const int e = tid / 12;
    const int c = (tid % 12) * 8;
    const int s = src[e], d = dst[e];
    const float norm = dinv[s] * w[e] * dinv[d];
    const float4 h0 = *(const float4*)(h + (size_t)s * 96 + c);
    const float4 h1 = *(const float4*)(h + (size_t)s * 96 + c + 4);
    float* ap = agg + (size_t)d * 96 + c;
    atomicAdd(ap + 0, h0.x * norm);
    atomicAdd(ap + 1, h0.y * norm);
    atomicAdd(ap + 2, h0.z * norm);
    atomicAdd(ap + 3, h0.w * norm);
    atomicAdd(ap + 4, h1.x * norm);
    atomicAdd(ap + 5, h1.y * norm);
    atomicAdd(ap + 6, h1.z * norm);
    atomicAdd(ap + 7, h1.w * norm);
}

// agg = relu(agg + h*dinv^2 + bias)   (in place on agg)
__global__ void combine_relu(float* __restrict__ agg, const float* __restrict__ h,
                             const float* __restrict__ dinv, const float* __restrict__ bias,
                             int total) {
    int i = blockIdx.x * blockDim.x + threadIdx.x;
    if (i >= total) return;
    const int n = i / DH;
    const int d = i - n * DH;
    const float di = dinv[n];
    const float v = agg[i] + h[i] * di * di + bias[d];
    agg[i] = fmaxf(v, 0.0f);
}

// pooled[batch[n]] += h[n]; cnts[g] += 1; first[g] = min(n)
__global__ void pool_kernel(const float* __restrict__ h, const int* __restrict__ batch,
                            float* __restrict__ pooled, float* __restrict__ cnts,
                            int* __restrict__ first, int total) {
    int i = blockIdx.x * blockDim.x + threadIdx.x;
    if (i >= total) return;
    const int n = i / DH;
    const int d = i - n * DH;
    const int g = batch[n];
    atomicAdd(&pooled[(size_t)g * DH + d], h[i]);
    if (d == 0) {
        atomicAdd(&cnts[g], 1.0f);
        atomicMin(&first[g], n);
    }
}

// out[g,o] = concat(mean_pool, relu(md@Wm+bm)) @ Wf + bf
__global__ void final_kernel(const float* __restrict__ pooled, const float* __restrict__ cnts,
                             const int* __restrict__ first, const float* __restrict__ metadata,
                             const float* __restrict__ Wm, const float* __restrict__ bm,
                             const float* __restrict__ Wf, const float* __restrict__ bf,
                             float* __restrict__ out) {
    int t = blockIdx.x * blockDim.x + threadIdx.x;
    if (t >= GG * DOUT) return;
    const int g = t / DOUT;
    const int o = t - g * DOUT;
    const float inv = 1.0f / fmaxf(cnts[g], 1.0f);
    const float* md = metadata + (size_t)(first[g] % GG) * MD;

    float acc = bf[o];
    for (int d = 0; d < DH; ++d)
        acc += pooled[(size_t)g * DH + d] * inv * Wf[d * DOUT + o];
    for (int j = 0; j < DH; ++j) {
        float mh = bm[j];
        for (int k = 0; k < MD; ++k) mh += md[k] * Wm[k * DH + j];
        mh = fmaxf(mh, 0.0f);
        acc += mh * Wf[(DH + j) * DOUT + o];
    }
    out[t] = acc;
}

// ---------------- launch ----------------
extern "C" void kernel_launch(void* const* d_in, const int* in_sizes, int n_in,
                              void* d_out, int out_size, void* d_ws, size_t ws_size,
                              hipStream_t stream) {
    (void)in_sizes; (void)n_in; (void)out_size; (void)ws_size;

    const float* x        = (const float*)d_in[0];
    const int*   ei       = (const int*)  d_in[1];   // [2,E]: src = ei, dst = ei+E
    const float* ew       = (const float*)d_in[2];
    const int*   batch    = (const int*)  d_in[3];
    const float* metadata = (const float*)d_in[4];
    const float* W1       = (const float*)d_in[5];
    const float* b1       = (const float*)d_in[6];
    const float* W2       = (const float*)d_in[7];
    const float* b2       = (const float*)d_in[8];
    const float* Wm       = (const float*)d_in[9];
    const float* bm       = (const float*)d_in[10];
    const float* Wf       = (const float*)d_in[11];
    const float* bf       = (const float*)d_in[12];
    float* out = (float*)d_out;

    // workspace layout (floats): dinv[50048 pad] | bufH[N*DH] | bufA[N*DH] | pooled[G*DH] | cnts[G] | first[G]
    float* ws     = (float*)d_ws;
    float* dinv   = ws;
    float* bufH   = ws + 50048;                 // 256B-aligned offset
    float* bufA   = bufH + (size_t)NN * DH;
    float* pooled = bufA + (size_t)NN * DH;
    float* cnts   = pooled + GG * DH;
    int*   first  = (int*)(cnts + GG);

    const int ND   = NN * DH;                    // 4,800,000
    const int TPB  = 256;
    const int tilesPerBlk = 4;                   // 128 threads = 4 waves
    const int rowTiles = NN / 16;                // 3125 (exact)
    const int gemmBlocks = (rowTiles + tilesPerBlk - 1) / tilesPerBlk;

    const int* src = ei;
    const int* dst = ei + EE;

    // degree -> dinv
    zero_f32<<<(NN + TPB - 1) / TPB, TPB, 0, stream>>>(dinv, NN);
    pool_init<<<(GG * DH + TPB - 1) / TPB, TPB, 0, stream>>>(pooled, cnts, first);
    deg_kernel<<<(EE + TPB - 1) / TPB, TPB, 0, stream>>>(dst, ew, dinv, EE);
    rsqrt_kernel<<<(NN + TPB - 1) / TPB, TPB, 0, stream>>>(dinv, NN);

    // ---- layer 1 ----
    gemm96_wmma<<<gemmBlocks, 32 * tilesPerBlk, 0, stream>>>(x, W1, bufH, NN);
    zero_f32<<<(ND + TPB - 1) / TPB, TPB, 0, stream>>>(bufA, ND);
    edge_agg<<<(EE * 12 + TPB - 1) / TPB, TPB, 0, stream>>>(src, dst, ew, dinv, bufH, bufA);
    combine_relu<<<(ND + TPB - 1) / TPB, TPB, 0, stream>>>(bufA, bufH, dinv, b1, ND);

    // ---- layer 2 ----
    gemm96_wmma<<<gemmBlocks, 32 * tilesPerBlk, 0, stream>>>(bufA, W2, bufH, NN);
    zero_f32<<<(ND + TPB - 1) / TPB, TPB, 0, stream>>>(bufA, ND);
    edge_agg<<<(EE * 12 + TPB - 1) / TPB, TPB, 0, stream>>>(src, dst, ew, dinv, bufH, bufA);
    combine_relu<<<(ND + TPB - 1) / TPB, TPB, 0, stream>>>(bufA, bufH, dinv, b2, ND);

    // ---- pool + heads ----
    pool_kernel<<<(ND + TPB - 1) / TPB, TPB, 0, stream>>>(bufA, batch, pooled, cnts, first, ND);
    final_kernel<<<1, GG * DOUT, 0, stream>>>(pooled, cnts, first, metadata, Wm, bm, Wf, bf, out);
}